// MuReNNLayer_56418690400558
// MI455X (gfx1250) — compile-verified
//
#include <hip/hip_runtime.h>
#include <math.h>

// ---------------- problem constants ----------------
#define N_SIG 65536   // samples per (b,c) signal
#define NB 4
#define NC 4
#define NQ 8
#define NT 15
#define NJ 5

// ---------------- fixed wavelet filters (compile-time constants) ----------------
__device__ const float cH0O[13] = {
  -0.00455690456024f, -0.00543947593727f,  0.01702522388155f,  0.02382538479492f,
  -0.10671180468666f,  0.01186609203379f,  0.56881042071212f,  0.75614564389252f,
   0.27529538466888f, -0.11720388769911f, -0.03887280126882f,  0.03466034684485f,
  -0.00388321199915f };
// H1O[t] = (-1)^t * H0O[12-t]
__device__ const float cH1O[13] = {
  -0.00388321199915f, -0.03466034684485f, -0.03887280126882f,  0.11720388769911f,
   0.27529538466888f, -0.75614564389252f,  0.56881042071212f, -0.01186609203379f,
  -0.10671180468666f, -0.02382538479492f,  0.01702522388155f,  0.00543947593727f,
  -0.00455690456024f };
__device__ const float cH0A[10] = {
   0.03516384f, 0.0f, -0.08832942f, 0.23389032f, 0.76027237f,
   0.58751830f, 0.0f, -0.11430184f, 0.0f, 0.0f };
__device__ const float cH0B[10] = {
   0.0f, 0.0f, -0.11430184f, 0.0f, 0.58751830f,
   0.76027237f, 0.23389032f, -0.08832942f, 0.0f, 0.03516384f };
__device__ const float cH1A[10] = {
   0.0f, 0.0f, -0.11430184f, 0.0f, 0.58751830f,
  -0.76027237f, 0.23389032f, 0.08832942f, 0.0f, -0.03516384f };
__device__ const float cH1B[10] = {
  -0.03516384f, 0.0f, 0.08832942f, 0.23389032f, -0.76027237f,
   0.58751830f, 0.0f, -0.11430184f, 0.0f, 0.0f };

typedef __attribute__((ext_vector_type(2))) float v2f;
typedef __attribute__((ext_vector_type(8))) float v8f;

// ---------------- K1: one udtcwt level -------------------------------------
// Reads la_in/lb_in (16 signals of 65536), writes la_out/lb_out and both
// bandpass trees (bpr from tree a, bpi from tree b) before the lowpass update.
// Each 256-thread block covers 256 consecutive n within one signal, so the
// interior/boundary split is block-uniform (scalar branch, no exec juggling).
__global__ void k_wavelet(const float* __restrict__ la_in, const float* __restrict__ lb_in,
                          float* __restrict__ la_out, float* __restrict__ lb_out,
                          float* __restrict__ bpr, float* __restrict__ bpi,
                          int level, int d)
{
  int idx = blockIdx.x * blockDim.x + threadIdx.x;     // 16 * 65536 threads
  int sig = idx >> 16;
  int n   = idx & (N_SIG - 1);
  int n_base = (blockIdx.x * blockDim.x) & (N_SIG - 1);
  const float* A  = la_in + (size_t)sig * N_SIG;
  const float* Bp = lb_in + (size_t)sig * N_SIG;
  float accLa = 0.f, accLb = 0.f, accBa = 0.f, accBb = 0.f;
  if (level == 0) {
    // pad = 6, max forward offset = 6
    if (n_base >= 6 && n_base + 255 + 6 < N_SIG) {
      const float* a = A  + n - 6;
      const float* bb = Bp + n - 6;
#pragma unroll
      for (int t = 0; t < 13; ++t) {
        float xa = a[t], xb = bb[t];
        accLa = fmaf(cH0O[t], xa, accLa);
        accBa = fmaf(cH1O[t], xa, accBa);
        accLb = fmaf(cH0O[t], xb, accLb);
        accBb = fmaf(cH1O[t], xb, accBb);
      }
    } else {
      int base = n - 6;
#pragma unroll
      for (int t = 0; t < 13; ++t) {
        int p = base + t;
        float xa = (p >= 0 && p < N_SIG) ? A[p]  : 0.f;
        float xb = (p >= 0 && p < N_SIG) ? Bp[p] : 0.f;
        accLa = fmaf(cH0O[t], xa, accLa);
        accBa = fmaf(cH1O[t], xa, accBa);
        accLb = fmaf(cH0O[t], xb, accLb);
        accBb = fmaf(cH1O[t], xb, accBb);
      }
    }
  } else {
    int pad = (9 * d) >> 1;                            // floor(d*(10-1)/2)
    int maxoff = 9 * d - pad;
    if (n_base >= pad && n_base + 255 + maxoff < N_SIG) {
      const float* a  = A  + n - pad;
      const float* bb = Bp + n - pad;
#pragma unroll
      for (int t = 0; t < 10; ++t) {
        float xa = a[d * t], xb = bb[d * t];
        accLa = fmaf(cH0A[t], xa, accLa);
        accBa = fmaf(cH1A[t], xa, accBa);
        accLb = fmaf(cH0B[t], xb, accLb);
        accBb = fmaf(cH1B[t], xb, accBb);
      }
    } else {
#pragma unroll
      for (int t = 0; t < 10; ++t) {
        int p = n + d * t - pad;
        float xa = (p >= 0 && p < N_SIG) ? A[p]  : 0.f;
        float xb = (p >= 0 && p < N_SIG) ? Bp[p] : 0.f;
        accLa = fmaf(cH0A[t], xa, accLa);
        accBa = fmaf(cH1A[t], xa, accBa);
        accLb = fmaf(cH0B[t], xb, accLb);
        accBb = fmaf(cH1B[t], xb, accBb);
      }
    }
  }
  la_out[idx] = accLa;  lb_out[idx] = accLb;
  bpr[idx]    = accBa;  bpi[idx]    = accBb;
}

// ---------------- K2: grouped conv via V_WMMA_F32_16X16X4_F32 + mod + pow ----
// A (16x4 f32): rows = q (8 real rows, 8 zero), K = 4-tap chunk of the 15 taps.
// B (4x16 f32): columns = 16 consecutive output positions, K = same tap chunk.
// D rows 0..7 live in lanes 0..15, VGPRs 0..7.
// Boundary handling: the in-range test depends only on n0 (wave-uniform), so
// interior tiles take a straight-line unguarded load path. Both paths fill the
// same registers and reconverge before the WMMA sequence (EXEC all-1s there).
__global__ void k_gconv_wmma(const float* __restrict__ bpr, const float* __restrict__ bpi,
                             const float* __restrict__ conv_w, const float* __restrict__ roots,
                             const float* __restrict__ beta_p,
                             float* __restrict__ u_out, int j, int d, float inv_scale)
{
  const int b     = blockIdx.x >> 2;
  const int c     = blockIdx.x & 3;
  const int wave  = threadIdx.x >> 5;
  const int lane  = threadIdx.x & 31;
  const int col   = lane & 15;        // M (for A) or N (for B) index
  const int khalf = lane >> 4;        // 0: K={4kc,4kc+1}, 1: K={4kc+2,4kc+3}

  // --- A matrix: learnable weights for this (j, c), taps padded 15 -> 16
  const float* W = conv_w + (size_t)(j * NC * NQ + c * NQ) * NT;   // W[q*15 + t]
  v2f aw[4];
#pragma unroll
  for (int kc = 0; kc < 4; ++kc) {
    int t0 = 4 * kc + 2 * khalf;
    float w0 = 0.f, w1 = 0.f;
    if (col < NQ) {
      if (t0     < NT) w0 = W[col * NT + t0];
      if (t0 + 1 < NT) w1 = W[col * NT + t0 + 1];
    }
    v2f a; a.x = w0; a.y = w1; aw[kc] = a;
  }

  // --- per-q exponents alpha = sigmoid(roots)
  float alpha[NQ];
#pragma unroll
  for (int q = 0; q < NQ; ++q) {
    float r = roots[j * NC * NQ + c * NQ + q];
    alpha[q] = 1.f / (1.f + __expf(-r));
  }
  const float beta = beta_p[0];

  const float* br_sig = bpr + (size_t)(b * NC + c) * N_SIG;
  const float* bi_sig = bpi + (size_t)(b * NC + c) * N_SIG;
  float* u_base = u_out + (size_t)(b * NC * NQ + c * NQ) * N_SIG;

  const int tiles  = N_SIG / 16;
  const int stride = gridDim.y * 8;
  for (int tile = blockIdx.y * 8 + wave; tile < tiles; tile += stride) {
    int n0 = tile * 16;
    // prefetch a future tile's window (lowers to global_prefetch_b8)
    __builtin_prefetch(br_sig + ((n0 + stride * 16) & (N_SIG - 1)), 0, 0);

    v2f brv[4], biv[4];
    // wave-uniform interior test: min index n0-7d, max index n0+15+8d
    if (n0 >= 7 * d && n0 + 15 + 8 * d < N_SIG) {
#pragma unroll
      for (int kc = 0; kc < 4; ++kc) {
        int t0 = 4 * kc + 2 * khalf;
        const float* pr = br_sig + n0 + col + d * (t0 - 7);
        const float* pi = bi_sig + n0 + col + d * (t0 - 7);
        brv[kc].x = pr[0]; brv[kc].y = pr[d];
        biv[kc].x = pi[0]; biv[kc].y = pi[d];
      }
    } else {
#pragma unroll
      for (int kc = 0; kc < 4; ++kc) {
        int t0 = 4 * kc + 2 * khalf;
        int i0 = n0 + col + d * (t0 - 7);
        int i1 = i0 + d;
        brv[kc].x = (i0 >= 0 && i0 < N_SIG) ? br_sig[i0] : 0.f;
        brv[kc].y = (i1 >= 0 && i1 < N_SIG) ? br_sig[i1] : 0.f;
        biv[kc].x = (i0 >= 0 && i0 < N_SIG) ? bi_sig[i0] : 0.f;
        biv[kc].y = (i1 >= 0 && i1 < N_SIG) ? bi_sig[i1] : 0.f;
      }
    }

    v8f accR = {0.f,0.f,0.f,0.f,0.f,0.f,0.f,0.f};
    v8f accI = {0.f,0.f,0.f,0.f,0.f,0.f,0.f,0.f};
#pragma unroll
    for (int kc = 0; kc < 4; ++kc) {
      accR = __builtin_amdgcn_wmma_f32_16x16x4_f32(false, aw[kc], false, brv[kc],
                                                   (short)0, accR, false, false);
      accI = __builtin_amdgcn_wmma_f32_16x16x4_f32(false, aw[kc], false, biv[kc],
                                                   (short)0, accI, false, false);
    }

    if (lane < 16) {                      // rows 0..7 (q) live here
#pragma unroll
      for (int q = 0; q < NQ; ++q) {
        float rr = accR[q], ii = accI[q];
        float u = sqrtf(fmaf(rr, rr, ii * ii)) * inv_scale;  // modulus / sqrt(2)^j
        u = fmaxf(u, -beta) + beta;                          // PowerStable shift
        u = powf(u, alpha[q]);
        u_base[(size_t)q * N_SIG + n0 + col] = u;
      }
    }
  }
}

// ---------------- K3: two decimating 13-tap lowpass stages -------------------
__global__ void k_down1(const float* __restrict__ u, float* __restrict__ y1)
{
  int idx = blockIdx.x * blockDim.x + threadIdx.x;  // 128 * 32768
  int sig = idx >> 15;
  int p   = idx & 32767;
  int p_base = (blockIdx.x * blockDim.x) & 32767;
  const float* s = u + (size_t)sig * N_SIG;
  float acc = 0.f;
  if (2 * p_base >= 6 && 2 * (p_base + 255) + 6 < N_SIG) {
    const float* sp = s + 2 * p - 6;
#pragma unroll
    for (int t = 0; t < 13; ++t) acc = fmaf(cH0O[t], sp[t], acc);
  } else {
    int base = 2 * p - 6;
#pragma unroll
    for (int t = 0; t < 13; ++t) {
      int q = base + t;
      float v = (q >= 0 && q < N_SIG) ? s[q] : 0.f;
      acc = fmaf(cH0O[t], v, acc);
    }
  }
  y1[idx] = acc;
}

__global__ void k_down2(const float* __restrict__ y1, float* __restrict__ out, int j)
{
  int idx = blockIdx.x * blockDim.x + threadIdx.x;  // 128 * 16384
  int sig = idx >> 14;                              // b*32 + o
  int m   = idx & 16383;
  int m_base = (blockIdx.x * blockDim.x) & 16383;
  int b   = sig >> 5;
  int o   = sig & 31;
  const float* s = y1 + (size_t)sig * 32768;
  float acc = 0.f;
  if (2 * m_base >= 6 && 2 * (m_base + 255) + 6 < 32768) {
    const float* sp = s + 2 * m - 6;
#pragma unroll
    for (int t = 0; t < 13; ++t) acc = fmaf(cH0O[t], sp[t], acc);
  } else {
    int base = 2 * m - 6;
#pragma unroll
    for (int t = 0; t < 13; ++t) {
      int q = base + t;
      float v = (q >= 0 && q < 32768) ? s[q] : 0.f;
      acc = fmaf(cH0O[t], v, acc);
    }
  }
  out[((size_t)b * (NJ * NC * NQ) + (size_t)j * NC * NQ + o) * 16384 + m] = acc;
}

// ---------------- host orchestration ----------------------------------------
extern "C" void kernel_launch(void* const* d_in, const int* in_sizes, int n_in,
                              void* d_out, int out_size, void* d_ws, size_t ws_size,
                              hipStream_t stream) {
  const float* x      = (const float*)d_in[0];   // (4,4,65536)
  const float* conv_w = (const float*)d_in[1];   // (5,32,1,15)
  const float* roots  = (const float*)d_in[2];   // (5,32)
  const float* beta   = (const float*)d_in[3];   // (1,)
  float* out = (float*)d_out;                    // (4,160,16384)
  float* ws  = (float*)d_ws;

  const size_t SIG = (size_t)16 * N_SIG;         // 1,048,576 floats (one la/lb set)
  // Workspace layout (floats): la0 lb0 la1 lb1 bpr bpi | u (8*SIG) | y1 (4*SIG)
  float* la0 = ws + 0 * SIG;
  float* lb0 = ws + 1 * SIG;
  float* la1 = ws + 2 * SIG;
  float* lb1 = ws + 3 * SIG;
  float* bpr = ws + 4 * SIG;
  float* bpi = ws + 5 * SIG;
  float* u   = ws + 6 * SIG;                     // 4*32*65536 = 8*SIG
  float* y1  = ws + 14 * SIG;                    // 4*32*32768 = 4*SIG
  (void)in_sizes; (void)n_in; (void)out_size; (void)ws_size;

  const float* curA = x;
  const float* curB = x;
  float* nla = la0; float* nlb = lb0;
  float* ola = la1; float* olb = lb1;

  for (int j = 0; j < NJ; ++j) {
    int d = (j == 0) ? 1 : (1 << (j - 1));
    float inv_scale = exp2f(-0.5f * (float)j);   // 1/sqrt(2)^j

    k_wavelet<<<4096, 256, 0, stream>>>(curA, curB, nla, nlb, bpr, bpi,
                                        (j == 0) ? 0 : 1, d);
    dim3 g2(16, 64);
    k_gconv_wmma<<<g2, 256, 0, stream>>>(bpr, bpi, conv_w, roots, beta,
                                         u, j, d, inv_scale);
    k_down1<<<16384, 256, 0, stream>>>(u, y1);
    k_down2<<<8192, 256, 0, stream>>>(y1, out, j);

    curA = nla; curB = nlb;
    float* t;
    t = nla; nla = ola; ola = t;
    t = nlb; nlb = olb; olb = t;
  }
}